// Router_1503238553992
// MI455X (gfx1250) — compile-verified
//
#include <hip/hip_runtime.h>

#define NUM_EXPERTS 8
#define C_DIM 2048
#define GATE_STRIDE 2056   // 2048 + 8 dword pad: row stride % 64 banks == 8 -> conflict-free
#define TOKENS_PER_WAVE 16
#define WAVES_PER_BLOCK 8
#define TOKENS_PER_BLOCK (TOKENS_PER_WAVE * WAVES_PER_BLOCK) // 128

typedef __attribute__((ext_vector_type(2))) float v2f;
typedef __attribute__((ext_vector_type(8))) float v8f;

// One wave computes a 16-token x 16-expert logits tile (experts 8..15 zero-padded)
// with V_WMMA_F32_16X16X4_F32, walking K=2048 in steps of 4.
// gate_w (64 KB) is staged once per block into LDS so the hot loop issues only
// one VMEM (x stream) + one DS (gate) per WMMA.
__global__ __launch_bounds__(256) void moe_router_kernel(
    const float* __restrict__ x, const float* __restrict__ gate_w,
    float* __restrict__ out, float* __restrict__ ws, int N)
{
  __shared__ float gsh[NUM_EXPERTS * GATE_STRIDE];            // padded gate, 64.25 KB
  __shared__ float vals[WAVES_PER_BLOCK][TOKENS_PER_WAVE][9]; // logits, then probs
  __shared__ float msk [WAVES_PER_BLOCK][TOKENS_PER_WAVE][9]; // top-2 multi-hot mask

  const int tid  = threadIdx.x;
  const int wave = tid >> 5;
  const int lane = tid & 31;
  const int n    = lane & 15;   // A: token row within tile; B: expert column
  const int half = lane >> 4;   // 0 or 1
  const int ksel = half * 2;    // K-pair select within the 4-wide K step

  // ---- Stage gate_w into LDS (vectorized, coalesced), rows padded ----
  {
    const float4* g4 = reinterpret_cast<const float4*>(gate_w);
    #pragma unroll
    for (int i = tid; i < (NUM_EXPERTS * C_DIM) / 4; i += 256) {
      const int r = i >> 9;            // / (2048/4)
      const int c = (i & 511) << 2;
      *reinterpret_cast<float4*>(&gsh[r * GATE_STRIDE + c]) = g4[i];
    }
  }
  __syncthreads();

  const int tokBase = blockIdx.x * TOKENS_PER_BLOCK + wave * TOKENS_PER_WAVE;

  // A fragment source: row (token) n of x
  const float* xrow = x + (size_t)(tokBase + n) * C_DIM;
  // B fragment source: LDS gate row (n&7); zero out padded experts 8..15
  const float  bscale = (n < NUM_EXPERTS) ? 1.0f : 0.0f;
  const float* grow   = &gsh[(n & 7) * GATE_STRIDE];

  v8f acc = {0.f, 0.f, 0.f, 0.f, 0.f, 0.f, 0.f, 0.f};

  #pragma unroll 8
  for (int k = 0; k < C_DIM; k += 4) {
    v2f a = *reinterpret_cast<const v2f*>(xrow + k + ksel);   // global_load_b64
    v2f b = *reinterpret_cast<const v2f*>(grow + k + ksel);   // ds_load_b64
    b.x *= bscale;
    b.y *= bscale;
    // D = A(16x4) * B(4x16) + C  -> v_wmma_f32_16x16x4_f32
    acc = __builtin_amdgcn_wmma_f32_16x16x4_f32(
        /*neg_a=*/false, a, /*neg_b=*/false, b,
        /*c_mod=*/(short)0, acc, /*reuse_a=*/false, /*reuse_b=*/false);
  }

  // C/D layout: lane L, reg r holds D[M = r + 8*(L>>4)][Nexp = L&15].
  // Scatter the 8 valid expert columns into LDS as [token][expert].
  if (n < NUM_EXPERTS) {
    #pragma unroll
    for (int r = 0; r < 8; ++r)
      vals[wave][r + 8 * half][n] = acc[r];
  }
  // Same-wave LDS ops are in-order; no barrier needed before same-wave reads.

  // Per-token epilogue: lanes 0..15 each own one token of this wave's tile.
  if (lane < TOKENS_PER_WAVE) {
    float l[NUM_EXPERTS];
    float m = -3.4e38f;
    #pragma unroll
    for (int e = 0; e < NUM_EXPERTS; ++e) {
      l[e] = vals[wave][lane][e];
      m = fmaxf(m, l[e]);
    }
    float p[NUM_EXPERTS];
    float s = 0.f;
    #pragma unroll
    for (int e = 0; e < NUM_EXPERTS; ++e) { p[e] = __expf(l[e] - m); s += p[e]; }
    const float inv = 1.0f / s;
    #pragma unroll
    for (int e = 0; e < NUM_EXPERTS; ++e) p[e] *= inv;

    // top-2, lowest index wins ties (matches jax.lax.top_k)
    int i0 = 0; float p0 = p[0];
    #pragma unroll
    for (int e = 1; e < NUM_EXPERTS; ++e) if (p[e] > p0) { p0 = p[e]; i0 = e; }
    int i1 = -1; float p1 = -1.f;
    #pragma unroll
    for (int e = 0; e < NUM_EXPERTS; ++e)
      if (e != i0 && p[e] > p1) { p1 = p[e]; i1 = e; }

    const float wsum = p0 + p1;
    const int g = tokBase + lane;
    out[2 * g]         = p0 / wsum;          // weights [N,2]
    out[2 * g + 1]     = p1 / wsum;
    out[2 * N + 2 * g]     = (float)i0;      // indices [N,2] (flattened as f32)
    out[2 * N + 2 * g + 1] = (float)i1;

    #pragma unroll
    for (int e = 0; e < NUM_EXPERTS; ++e) {
      vals[wave][lane][e] = p[e];
      msk [wave][lane][e] = (e == i0 || e == i1) ? 1.0f : 0.0f;
    }
  }

  __syncthreads();

  // Deterministic per-block column sums: tid<8 -> sum probs, tid 8..15 -> sum mask.
  if (tid < 16) {
    const int e = tid & 7;
    const bool isMask = tid >= 8;
    float s = 0.f;
    for (int t = 0; t < TOKENS_PER_BLOCK; ++t) {
      const int wv = t >> 4, tt = t & 15;
      s += isMask ? msk[wv][tt][e] : vals[wv][tt][e];
    }
    ws[blockIdx.x * 16 + tid] = s;
  }
}

__global__ void moe_finalize_kernel(const float* __restrict__ ws,
                                    float* __restrict__ out,
                                    int numBlocks, int N)
{
  __shared__ float tot[16];
  const int tid = threadIdx.x;
  if (tid < 16) {
    float s = 0.f;
    for (int b = 0; b < numBlocks; ++b) s += ws[b * 16 + tid];
    tot[tid] = s;
  }
  __syncthreads();
  if (tid == 0) {
    const float invN = 1.0f / (float)N;
    float aux = 0.f;
    for (int e = 0; e < NUM_EXPERTS; ++e)
      aux += (tot[8 + e] * invN) * (tot[e] * invN);   // f[e] * P[e]
    out[4 * N] = (float)NUM_EXPERTS * aux;            // aux_loss scalar
  }
}

extern "C" void kernel_launch(void* const* d_in, const int* in_sizes, int n_in,
                              void* d_out, int out_size, void* d_ws, size_t ws_size,
                              hipStream_t stream) {
  const float* x      = (const float*)d_in[0];  // [4,4096,2048] f32
  const float* gate_w = (const float*)d_in[1];  // [8,2048] f32
  float* out = (float*)d_out;                   // weights(2N) | indices(2N) | aux(1)
  float* ws  = (float*)d_ws;                    // numBlocks*16 partial sums

  const int N = in_sizes[0] / C_DIM;            // 16384 tokens
  const int numBlocks = N / TOKENS_PER_BLOCK;   // 128

  moe_router_kernel<<<numBlocks, 256, 0, stream>>>(x, gate_w, out, ws, N);
  moe_finalize_kernel<<<1, 32, 0, stream>>>(ws, out, numBlocks, N);
}